// HAN_59493886984413
// MI455X (gfx1250) — compile-verified
//
#include <hip/hip_runtime.h>

#define NU 100000
#define NG 50000
#define EE 1000000
#define CC 64
#define HH 8
#define NEG_SLOPE 0.2f

typedef __attribute__((ext_vector_type(2))) float v2f;
typedef __attribute__((ext_vector_type(8))) float v8f;

// ---- monotonic float<->uint mapping so atomicMax(uint) == float max ----
__device__ __forceinline__ unsigned enc_f(float f) {
  unsigned b = __float_as_uint(f);
  return (b & 0x80000000u) ? ~b : (b | 0x80000000u);
}
__device__ __forceinline__ float dec_f(unsigned u) {
  unsigned b = (u & 0x80000000u) ? (u & 0x7fffffffu) : ~u;
  return __uint_as_float(b);
}
__device__ __forceinline__ float lrelu(float a) { return a >= 0.f ? a : NEG_SLOPE * a; }

// ============================================================
// Y[N,64] = X[N,64] @ W[64,64]^T + B   via V_WMMA_F32_16X16X4_F32
// Block = 256 threads = 8 waves; each wave owns a 16x16 tile of Y.
// A layout: lane&15 = M, lane>>4 picks K pair {0,1}/{2,3} of the K=4 slab.
// B layout: lane&15 = N (output channel), same K split.
// D layout: lane>>4 picks rows 0-7 / 8-15 in the 8 accumulator VGPRs.
// ============================================================
__global__ void linear64_wmma(const float* __restrict__ X, const float* __restrict__ W,
                              const float* __restrict__ B, float* __restrict__ Y, int N) {
  const int lane = threadIdx.x & 31;
  const int wave = threadIdx.x >> 5;
  const int row_base = blockIdx.x * 32 + (wave >> 2) * 16;
  const int col_base = (wave & 3) * 16;
  const int mn = lane & 15;
  const int khalf = lane >> 4;
  int arow = row_base + mn;
  if (arow >= N) arow = N - 1;  // clamp loads; stores predicated below
  const float* xrow = X + (size_t)arow * CC;
  const float* wrow = W + (size_t)(col_base + mn) * CC;
  v8f acc = {};
#pragma unroll
  for (int k0 = 0; k0 < CC; k0 += 4) {
    v2f a = *(const v2f*)(xrow + k0 + 2 * khalf);
    v2f b = *(const v2f*)(wrow + k0 + 2 * khalf);
    acc = __builtin_amdgcn_wmma_f32_16x16x4_f32(false, a, false, b, (short)0, acc, false, false);
  }
  const float bias = B[col_base + mn];
  const int mrow0 = row_base + khalf * 8;
#pragma unroll
  for (int v = 0; v < 8; v++) {
    int r = mrow0 + v;
    if (r < N) Y[(size_t)r * CC + col_base + mn] = acc[v] + bias;
  }
}

// ============================================================
// colsum[c] += sum_n tanh( (relu(O) @ KW^T + KB)[n][c] )   (semantic attn GEMM)
// ============================================================
__global__ void tanh_colsum_wmma(const float* __restrict__ O, const float* __restrict__ KW,
                                 const float* __restrict__ KB, float* __restrict__ colsum, int N) {
  const int lane = threadIdx.x & 31;
  const int wave = threadIdx.x >> 5;
  const int row_base = blockIdx.x * 32 + (wave >> 2) * 16;
  const int col_base = (wave & 3) * 16;
  const int mn = lane & 15;
  const int khalf = lane >> 4;
  int arow = row_base + mn;
  if (arow >= N) arow = N - 1;
  const float* orow = O + (size_t)arow * CC;
  const float* wrow = KW + (size_t)(col_base + mn) * CC;
  v8f acc = {};
#pragma unroll
  for (int k0 = 0; k0 < CC; k0 += 4) {
    v2f a, b;
    a.x = fmaxf(orow[k0 + 2 * khalf], 0.f);      // relu fused into A load
    a.y = fmaxf(orow[k0 + 2 * khalf + 1], 0.f);
    b = *(const v2f*)(wrow + k0 + 2 * khalf);
    acc = __builtin_amdgcn_wmma_f32_16x16x4_f32(false, a, false, b, (short)0, acc, false, false);
  }
  const float kb = KB[col_base + mn];
  float part = 0.f;
  const int mrow0 = row_base + khalf * 8;
#pragma unroll
  for (int v = 0; v < 8; v++) {
    int r = mrow0 + v;
    if (r < N) part += tanhf(acc[v] + kb);
  }
  atomicAdd(&colsum[col_base + mn], part);
}

// ============================================================
// Per-node attention logits: o_j[node,h] = sum_d H[node,h,d] * a_j[h,d]
// ============================================================
__global__ void node_alpha(const float* __restrict__ Hf, int N,
                           const float* __restrict__ a0, const float* __restrict__ a1,
                           const float* __restrict__ a2, const float* __restrict__ a3,
                           float* __restrict__ o0, float* __restrict__ o1,
                           float* __restrict__ o2, float* __restrict__ o3) {
  int idx = blockIdx.x * blockDim.x + threadIdx.x;
  if (idx >= N * HH) return;
  int node = idx >> 3, h = idx & 7;
  const float* hv = Hf + (size_t)node * CC + h * 8;
  float s0 = 0, s1 = 0, s2 = 0, s3 = 0;
#pragma unroll
  for (int d = 0; d < 8; d++) {
    float x = hv[d];
    s0 += x * a0[h * 8 + d];
    s1 += x * a1[h * 8 + d];
    s2 += x * a2[h * 8 + d];
    s3 += x * a3[h * 8 + d];
  }
  o0[idx] = s0; o1[idx] = s1; o2[idx] = s2; o3[idx] = s3;
}

// ---- segment softmax pass 1: per-(dst,head) running max ----
__global__ void edge_max(const int* __restrict__ src, const int* __restrict__ dst,
                         const float* __restrict__ als, const float* __restrict__ ald,
                         unsigned* __restrict__ menc) {
  int idx = blockIdx.x * blockDim.x + threadIdx.x;
  if (idx >= EE * HH) return;
  int e = idx >> 3, h = idx & 7;
  int s = src[e], d = dst[e];
  float a = lrelu(als[s * 8 + h] + ald[d * 8 + h]);
  atomicMax(&menc[d * 8 + h], enc_f(a));
}

// ---- pass 2: per-(dst,head) sum of exp(alpha - max) ----
__global__ void edge_sum(const int* __restrict__ src, const int* __restrict__ dst,
                         const float* __restrict__ als, const float* __restrict__ ald,
                         const unsigned* __restrict__ menc, float* __restrict__ ssum) {
  int idx = blockIdx.x * blockDim.x + threadIdx.x;
  if (idx >= EE * HH) return;
  int e = idx >> 3, h = idx & 7;
  int s = src[e], d = dst[e];
  float a = lrelu(als[s * 8 + h] + ald[d * 8 + h]);
  atomicAdd(&ssum[d * 8 + h], expf(a - dec_f(menc[d * 8 + h])));
}

// ---- pass 3: out[dst] += softmax_weight * h_src[src]  (64 lanes per edge) ----
__global__ void edge_scatter(const int* __restrict__ src, const int* __restrict__ dst,
                             const float* __restrict__ als, const float* __restrict__ ald,
                             const unsigned* __restrict__ menc, const float* __restrict__ ssum,
                             const float* __restrict__ Hsrc, float* __restrict__ outacc) {
  int idx = blockIdx.x * blockDim.x + threadIdx.x;
  if (idx >= EE * CC) return;
  int e = idx >> 6, c = idx & 63, h = c >> 3;
  int s = src[e], d = dst[e];
  float a = lrelu(als[s * 8 + h] + ald[d * 8 + h]);
  float ex = expf(a - dec_f(menc[d * 8 + h]));
  float w = ex / (ssum[d * 8 + h] + 1e-16f);
  atomicAdd(&outacc[(size_t)d * CC + c], w * Hsrc[(size_t)s * CC + c]);
}

// ---- semantic scores + softmax over the two metapaths per node type ----
__global__ void finalize_scores(const float* __restrict__ colsum, const float* __restrict__ q,
                                float* __restrict__ wsem) {
  float sc[4];
  const float invN[4] = {1.f / NU, 1.f / NU, 1.f / NG, 1.f / NG};
  for (int m = 0; m < 4; m++) {
    float acc = 0.f;
    for (int c = 0; c < CC; c++) acc += q[c] * colsum[m * CC + c] * invN[m];
    sc[m] = acc;
  }
  {
    float mx = fmaxf(sc[0], sc[1]);
    float e0 = expf(sc[0] - mx), e1 = expf(sc[1] - mx);
    wsem[0] = e0 / (e0 + e1); wsem[1] = e1 / (e0 + e1);
  }
  {
    float mx = fmaxf(sc[2], sc[3]);
    float e0 = expf(sc[2] - mx), e1 = expf(sc[3] - mx);
    wsem[2] = e0 / (e0 + e1); wsem[3] = e1 / (e0 + e1);
  }
}

__global__ void combine(const float* __restrict__ o0, const float* __restrict__ o1,
                        const float* __restrict__ wsem, int wofs,
                        float* __restrict__ out, int n) {
  int idx = blockIdx.x * blockDim.x + threadIdx.x;
  if (idx >= n) return;
  out[idx] = wsem[wofs] * fmaxf(o0[idx], 0.f) + wsem[wofs + 1] * fmaxf(o1[idx], 0.f);
}

extern "C" void kernel_launch(void* const* d_in, const int* in_sizes, int n_in,
                              void* d_out, int out_size, void* d_ws, size_t ws_size,
                              hipStream_t stream) {
  const float* x_user = (const float*)d_in[0];
  const float* x_game = (const float*)d_in[1];
  const int* plays_src = (const int*)d_in[2];
  const int* plays_dst = (const int*)d_in[3];
  const int* rec_src = (const int*)d_in[4];
  const int* rec_dst = (const int*)d_in[5];
  const float* W_user = (const float*)d_in[6];  const float* b_user = (const float*)d_in[7];
  const float* W_game = (const float*)d_in[8];  const float* b_game = (const float*)d_in[9];
  const float* att_src_pg = (const float*)d_in[10]; const float* att_dst_pg = (const float*)d_in[11];
  const float* att_src_rg = (const float*)d_in[12]; const float* att_dst_rg = (const float*)d_in[13];
  const float* att_src_pu = (const float*)d_in[14]; const float* att_dst_pu = (const float*)d_in[15];
  const float* att_src_ru = (const float*)d_in[16]; const float* att_dst_ru = (const float*)d_in[17];
  const float* k_lin_w = (const float*)d_in[18]; const float* k_lin_b = (const float*)d_in[19];
  const float* q = (const float*)d_in[20];

  float* ws = (float*)d_ws;
  size_t off = 0;
  auto alloc = [&](size_t n) { float* p = ws + off; off += n; return p; };
  float* h_u = alloc((size_t)NU * CC);
  float* h_g = alloc((size_t)NG * CC);
  float* al_u_s_pg = alloc((size_t)NU * HH);
  float* al_u_s_rg = alloc((size_t)NU * HH);
  float* al_u_d_pu = alloc((size_t)NU * HH);
  float* al_u_d_ru = alloc((size_t)NU * HH);
  float* al_g_d_pg = alloc((size_t)NG * HH);
  float* al_g_d_rg = alloc((size_t)NG * HH);
  float* al_g_s_pu = alloc((size_t)NG * HH);
  float* al_g_s_ru = alloc((size_t)NG * HH);
  size_t zero_start = off;  // everything below gets zeroed each launch
  unsigned* m_g_pg = (unsigned*)alloc((size_t)NG * HH);
  float*    s_g_pg = alloc((size_t)NG * HH);
  unsigned* m_g_rg = (unsigned*)alloc((size_t)NG * HH);
  float*    s_g_rg = alloc((size_t)NG * HH);
  unsigned* m_u_pu = (unsigned*)alloc((size_t)NU * HH);
  float*    s_u_pu = alloc((size_t)NU * HH);
  unsigned* m_u_ru = (unsigned*)alloc((size_t)NU * HH);
  float*    s_u_ru = alloc((size_t)NU * HH);
  float* og_p = alloc((size_t)NG * CC);
  float* og_r = alloc((size_t)NG * CC);
  float* ou_p = alloc((size_t)NU * CC);
  float* ou_r = alloc((size_t)NU * CC);
  float* colsum = alloc(4 * CC);
  float* wsem = alloc(4);
  hipMemsetAsync(ws + zero_start, 0, (off - zero_start) * sizeof(float), stream);

  const int TB = 256;
  // Dense linears (WMMA)
  linear64_wmma<<<(NU + 31) / 32, TB, 0, stream>>>(x_user, W_user, b_user, h_u, NU);
  linear64_wmma<<<(NG + 31) / 32, TB, 0, stream>>>(x_game, W_game, b_game, h_g, NG);
  // Per-node attention logits
  node_alpha<<<(NU * HH + TB - 1) / TB, TB, 0, stream>>>(
      h_u, NU, att_src_pg, att_src_rg, att_dst_pu, att_dst_ru,
      al_u_s_pg, al_u_s_rg, al_u_d_pu, al_u_d_ru);
  node_alpha<<<(NG * HH + TB - 1) / TB, TB, 0, stream>>>(
      h_g, NG, att_dst_pg, att_dst_rg, att_src_pu, att_src_ru,
      al_g_d_pg, al_g_d_rg, al_g_s_pu, al_g_s_ru);

  const int gbE8 = (EE * HH + TB - 1) / TB;
  const int gbE64 = (EE * CC + TB - 1) / TB;
  // plays: user -> game
  edge_max<<<gbE8, TB, 0, stream>>>(plays_src, plays_dst, al_u_s_pg, al_g_d_pg, m_g_pg);
  edge_sum<<<gbE8, TB, 0, stream>>>(plays_src, plays_dst, al_u_s_pg, al_g_d_pg, m_g_pg, s_g_pg);
  edge_scatter<<<gbE64, TB, 0, stream>>>(plays_src, plays_dst, al_u_s_pg, al_g_d_pg, m_g_pg, s_g_pg, h_u, og_p);
  // recommends: user -> game
  edge_max<<<gbE8, TB, 0, stream>>>(rec_src, rec_dst, al_u_s_rg, al_g_d_rg, m_g_rg);
  edge_sum<<<gbE8, TB, 0, stream>>>(rec_src, rec_dst, al_u_s_rg, al_g_d_rg, m_g_rg, s_g_rg);
  edge_scatter<<<gbE64, TB, 0, stream>>>(rec_src, rec_dst, al_u_s_rg, al_g_d_rg, m_g_rg, s_g_rg, h_u, og_r);
  // reversed plays: game -> user
  edge_max<<<gbE8, TB, 0, stream>>>(plays_dst, plays_src, al_g_s_pu, al_u_d_pu, m_u_pu);
  edge_sum<<<gbE8, TB, 0, stream>>>(plays_dst, plays_src, al_g_s_pu, al_u_d_pu, m_u_pu, s_u_pu);
  edge_scatter<<<gbE64, TB, 0, stream>>>(plays_dst, plays_src, al_g_s_pu, al_u_d_pu, m_u_pu, s_u_pu, h_g, ou_p);
  // reversed recommends: game -> user
  edge_max<<<gbE8, TB, 0, stream>>>(rec_dst, rec_src, al_g_s_ru, al_u_d_ru, m_u_ru);
  edge_sum<<<gbE8, TB, 0, stream>>>(rec_dst, rec_src, al_g_s_ru, al_u_d_ru, m_u_ru, s_u_ru);
  edge_scatter<<<gbE64, TB, 0, stream>>>(rec_dst, rec_src, al_g_s_ru, al_u_d_ru, m_u_ru, s_u_ru, h_g, ou_r);

  // Semantic attention (WMMA GEMM + tanh + column reduce); m0=ou_p m1=ou_r m2=og_p m3=og_r
  tanh_colsum_wmma<<<(NU + 31) / 32, TB, 0, stream>>>(ou_p, k_lin_w, k_lin_b, colsum + 0 * CC, NU);
  tanh_colsum_wmma<<<(NU + 31) / 32, TB, 0, stream>>>(ou_r, k_lin_w, k_lin_b, colsum + 1 * CC, NU);
  tanh_colsum_wmma<<<(NG + 31) / 32, TB, 0, stream>>>(og_p, k_lin_w, k_lin_b, colsum + 2 * CC, NG);
  tanh_colsum_wmma<<<(NG + 31) / 32, TB, 0, stream>>>(og_r, k_lin_w, k_lin_b, colsum + 3 * CC, NG);
  finalize_scores<<<1, 1, 0, stream>>>(colsum, q, wsem);

  float* out = (float*)d_out;
  combine<<<(NU * CC + TB - 1) / TB, TB, 0, stream>>>(ou_p, ou_r, wsem, 0, out, NU * CC);
  combine<<<(NG * CC + TB - 1) / TB, TB, 0, stream>>>(og_p, og_r, wsem, 2, out + (size_t)NU * CC, NG * CC);
}